// Invert_36129264894624
// MI455X (gfx1250) — compile-verified
//
#include <hip/hip_runtime.h>

typedef __attribute__((ext_vector_type(2))) float v2f;
typedef __attribute__((ext_vector_type(8))) float v8f;

// One wave32 workgroup. Lane 0 builds X_full = I + scatter(X) and inverts it
// via Gauss-Jordan (5x5, fully unrolled, stays in VGPRs). All 32 lanes then
// assemble V_WMMA_F32_16X16X4_F32 fragments from LDS and compute
// Y = Xinv * A with two K=4 WMMA steps (K padded 5 -> 8). Finally the
// sum((Y - I)^2) reduction runs over lanes 0..4 / D-VGPRs 0..4.
__global__ __launch_bounds__(32) void invert_loss_kernel(
    const float* __restrict__ X,   // 7 values
    const float* __restrict__ A,   // 5x5 row-major
    float* __restrict__ out) {     // 1 scalar
  __shared__ float sXinv[5][5];
  __shared__ float sA[5][5];
  __shared__ float sRed[8];

  const int lane = threadIdx.x;

  // ---- Serial 5x5 Gauss-Jordan inversion on lane 0 (registers only) ----
  if (lane == 0) {
    float M[5][10];
#pragma unroll
    for (int r = 0; r < 5; ++r)
#pragma unroll
      for (int c = 0; c < 10; ++c)
        M[r][c] = 0.0f;
#pragma unroll
    for (int r = 0; r < 5; ++r) { M[r][r] = 1.0f; M[r][5 + r] = 1.0f; }
    // sparse scatter: 5 diagonal entries + (0,4) and (4,0)
    M[0][0] += X[0]; M[1][1] += X[1]; M[2][2] += X[2];
    M[3][3] += X[3]; M[4][4] += X[4];
    M[0][4] += X[5]; M[4][0] += X[6];
#pragma unroll
    for (int p = 0; p < 5; ++p) {
      const float inv = 1.0f / M[p][p];
#pragma unroll
      for (int c = 0; c < 10; ++c) M[p][c] *= inv;
#pragma unroll
      for (int r = 0; r < 5; ++r) {
        if (r == p) continue;
        const float f = M[r][p];
#pragma unroll
        for (int c = 0; c < 10; ++c) M[r][c] -= f * M[p][c];
      }
    }
#pragma unroll
    for (int r = 0; r < 5; ++r)
#pragma unroll
      for (int c = 0; c < 5; ++c)
        sXinv[r][c] = M[r][5 + c];
  }
  if (lane < 25) sA[lane / 5][lane % 5] = A[lane];
  __syncthreads();   // single-wave group: barrier is S_NOP, but orders LDS

  // ---- Build WMMA fragments (EXEC is all-1s from here on) ----
  const int half = lane >> 4;      // 0: K-pair {0,1}, 1: K-pair {2,3}
  const int mrow = lane & 15;      // A-fragment row / B-fragment column
  const bool mLive = (mrow < 5);

  // A-matrix (Xinv), first WMMA: K = 0..3
  v2f a0;
  a0.x = mLive ? sXinv[mrow][2 * half + 0] : 0.0f;
  a0.y = mLive ? sXinv[mrow][2 * half + 1] : 0.0f;
  // A-matrix, second WMMA: K = 4..7 (only K=4 live, lanes 0-15 v0)
  v2f a1;
  a1.x = (mLive && half == 0) ? sXinv[mrow][4] : 0.0f;
  a1.y = 0.0f;

  // B-matrix (A), first WMMA: K = 0..3
  v2f b0;
  b0.x = mLive ? sA[2 * half + 0][mrow] : 0.0f;
  b0.y = mLive ? sA[2 * half + 1][mrow] : 0.0f;
  // B-matrix, second WMMA: K = 4..7 (only K=4 live)
  v2f b1;
  b1.x = (mLive && half == 0) ? sA[4][mrow] : 0.0f;
  b1.y = 0.0f;

  // ---- Y = Xinv * A via two accumulating fp32 WMMAs ----
  v8f acc = {};
  acc = __builtin_amdgcn_wmma_f32_16x16x4_f32(
      /*neg_a=*/false, a0, /*neg_b=*/false, b0,
      /*c_mod=*/(short)0, acc, /*reuse_a=*/false, /*reuse_b=*/false);
  acc = __builtin_amdgcn_wmma_f32_16x16x4_f32(
      /*neg_a=*/false, a1, /*neg_b=*/false, b1,
      /*c_mod=*/(short)0, acc, /*reuse_a=*/false, /*reuse_b=*/false);

  // ---- sum((Y - I)^2): lanes 0-4 hold columns 0-4; VGPR r holds row r ----
  float s = 0.0f;
  if (lane < 5) {
#pragma unroll
    for (int r = 0; r < 5; ++r) {
      const float d = acc[r] - ((r == lane) ? 1.0f : 0.0f);
      s += d * d;
    }
  }
  if (lane < 8) sRed[lane] = (lane < 5) ? s : 0.0f;
  __syncthreads();
  if (lane == 0) {
    float total = 0.0f;
#pragma unroll
    for (int i = 0; i < 5; ++i) total += sRed[i];
    out[0] = total;
  }
}

extern "C" void kernel_launch(void* const* d_in, const int* in_sizes, int n_in,
                              void* d_out, int out_size, void* d_ws, size_t ws_size,
                              hipStream_t stream) {
  (void)in_sizes; (void)n_in; (void)out_size; (void)d_ws; (void)ws_size;
  const float* X = (const float*)d_in[0];   // 7 sparse values
  const float* A = (const float*)d_in[1];   // 5x5
  float* out = (float*)d_out;               // 1 scalar
  invert_loss_kernel<<<1, 32, 0, stream>>>(X, A, out);
}